// Net_44925357916450
// MI455X (gfx1250) — compile-verified
//
#include <hip/hip_runtime.h>
#include <hip/hip_bf16.h>
#include <math.h>
#include <stdint.h>

typedef float v2f __attribute__((ext_vector_type(2)));
typedef float v8f __attribute__((ext_vector_type(8)));

// ---------------------------------------------------------------------------
// Kernel 1: per-batch mean objectness over anchors + stable top-32 selection.
// boxes: [bs, 169, 3, 5]; idxOut: [bs, 32]
// ---------------------------------------------------------------------------
__global__ void topk_kernel(const float* __restrict__ boxes, int* __restrict__ idxOut)
{
    const int G = 169, KSEL = 32;
    const int b = blockIdx.x;
    const int t = threadIdx.x;              // 256 threads

    __shared__ float score[192];
    __shared__ float rs[256];
    __shared__ int   ri[256];

    const float* bb = boxes + (size_t)b * G * 15;   // 3 anchors * 5 ch
    if (t < G) {
        score[t] = (bb[t * 15 + 4] + bb[t * 15 + 9] + bb[t * 15 + 14]) * (1.0f / 3.0f);
    }
    __syncthreads();

    for (int sel = 0; sel < KSEL; ++sel) {
        rs[t] = (t < G) ? score[t] : -INFINITY;
        ri[t] = t;
        __syncthreads();
        for (int off = 128; off >= 1; off >>= 1) {
            if (t < off) {
                float s2 = rs[t + off]; int i2 = ri[t + off];
                // max with tie -> lowest index (matches jax.lax.top_k stability)
                if (s2 > rs[t] || (s2 == rs[t] && i2 < ri[t])) { rs[t] = s2; ri[t] = i2; }
            }
            __syncthreads();
        }
        if (t == 0) {
            idxOut[b * KSEL + sel] = ri[0];
            score[ri[0]] = -INFINITY;
        }
        __syncthreads();
    }
}

// ---------------------------------------------------------------------------
// Kernel 2/3: fp32 GEMM via V_WMMA_F32_16X16X4_F32. One wave per 16x16 tile.
//   C[M,N] = A[M,K] @ B (+ bias)
//   BT==0 : B row-major [K,N];  BT==1 : B stored [N,K]
// Template params remove all runtime-uniform branches from the K-loop.
// fp32 WMMA VGPR layouts (ISA 7.12.2):
//   A 16x4 : lanes 0-15 -> M=lane, v0=K+0, v1=K+1 ; lanes 16-31 -> K+2, K+3
//   B 4x16 : lanes 0-15 -> N=lane, v0=K+0, v1=K+1 ; lanes 16-31 -> K+2, K+3
//   C/D    : VGPR r -> M = half*8 + r, N = lane&15
// ---------------------------------------------------------------------------
template <int BT, int HASBIAS>
__global__ void gemm_f32_wmma(const float* __restrict__ A, const float* __restrict__ B,
                              float* __restrict__ C, const float* __restrict__ bias,
                              int M, int N, int K)
{
    const int lane = threadIdx.x & 31;
    const int half = lane >> 4;       // 0: K{0,1}; 1: K{2,3}
    const int l16  = lane & 15;
    const int tn   = blockIdx.x;      // N/16 tiles
    const int tm   = blockIdx.y;      // M/16 tiles

    const int mRow = tm * 16 + l16;
    const int nCol = tn * 16 + l16;

    const float* arow = A + (size_t)mRow * K + half * 2;   // this lane's A K-pair base
    const float* bcol;
    if (BT) bcol = B + (size_t)nCol * K + half * 2;        // [N,K]: contiguous K pairs
    else    bcol = B + (size_t)(half * 2) * N + nCol;      // [K,N]: stride-N pairs

    v8f acc = {};
#pragma unroll 4
    for (int k = 0; k < K; k += 4) {
        v2f a = *(const v2f*)(arow + k);                   // global_load_b64
        __builtin_prefetch(arow + k + 64, 0, 3);           // global_prefetch_b8 (near)
        v2f bf;
        if (BT) {
            bf = *(const v2f*)(bcol + k);                  // global_load_b64
        } else {
            const float* bp = bcol + (size_t)k * N;
            bf.x = bp[0];
            bf.y = bp[N];
        }
        acc = __builtin_amdgcn_wmma_f32_16x16x4_f32(
            /*neg_a=*/false, a, /*neg_b=*/false, bf,
            /*c_mod=*/(short)0, acc, /*reuse_a=*/false, /*reuse_b=*/false);
    }

    float badd = 0.0f;
    if (HASBIAS) badd = bias[nCol];
#pragma unroll
    for (int r = 0; r < 8; ++r) {
        const int m = tm * 16 + half * 8 + r;
        C[(size_t)m * N + nCol] = acc[r] + badd;
    }
}

// ---------------------------------------------------------------------------
// Kernel 4: fused similarity + selection + box transform + output gather.
//   sim[b,k] = <feat[b,:,g_k], v[b,:]> + c[b],   c[b] = <b_vs, y_new[b]>
// One 256-thread block (8 waves) per batch. v[b,:] (4 KB) is staged into LDS
// once via async global->LDS (each of 256 lanes moves 16 B), then reused by
// all 32 dot products instead of 4x-redundant global reads.
// ---------------------------------------------------------------------------
__global__ void sim_select_kernel(const float* __restrict__ feat,   // [bs,1024,169]
                                  const float* __restrict__ vvec,   // [bs,1024]
                                  const float* __restrict__ y_new,  // [bs,512]
                                  const float* __restrict__ b_vs,   // [512]
                                  const int*   __restrict__ idx,    // [bs,32]
                                  const float* __restrict__ boxes,  // [bs,169,3,5]
                                  const float* __restrict__ masks,  // [bs,169,3,32]
                                  float* __restrict__ outBox,       // [bs,5]
                                  float* __restrict__ outMask,      // [bs,32]
                                  float* __restrict__ outMax)       // [bs]
{
    const int G = 169, D = 1024, H = 512, KSEL = 32;
    const int b    = blockIdx.x;
    const int t    = threadIdx.x;      // 256
    const int lane = t & 31;
    const int wv   = t >> 5;           // 8 waves

    __shared__ float vsm[1024];        // staged v[b,:]
    __shared__ float red[256];
    __shared__ float simS[32];
    __shared__ float cS;
    __shared__ int   gStar, aStar;

    const float* vb = vvec + (size_t)b * D;

    // --- async stage v[b,:] into LDS: 256 lanes x 16B = 4096B, one op/lane ---
    {
        const float* gp  = vb + t * 4;
        // low 32 bits of a flat pointer to LDS == LDS byte offset (ISA 10.2)
        uint32_t lds = (uint32_t)(uintptr_t)(&vsm[t * 4]);
        asm volatile("global_load_async_to_lds_b128 %0, %1, off"
                     :: "v"(lds), "v"(gp) : "memory");
    }

    // c[b] = dot(b_vs, y_new[b])  (overlaps with the async copy in flight)
    float p = 0.0f;
    for (int h = t; h < H; h += 256) p += b_vs[h] * y_new[(size_t)b * H + h];
    red[t] = p;

    asm volatile("s_wait_asynccnt 0" ::: "memory");
    __syncthreads();
    for (int off = 128; off >= 1; off >>= 1) {
        if (t < off) red[t] += red[t + off];
        __syncthreads();
    }
    if (t == 0) cS = red[0];
    __syncthreads();

    const float* fb = feat + (size_t)b * D * G;
    for (int k = wv; k < KSEL; k += 8) {
        const int g = idx[b * KSEL + k];
        float s = 0.0f;
#pragma unroll 4
        for (int d = lane; d < D; d += 32)
            s += fb[(size_t)d * G + g] * vsm[d];
        for (int off = 16; off >= 1; off >>= 1) s += __shfl_xor(s, off, 32);
        if (lane == 0) simS[k] = s + cS;
    }
    __syncthreads();

    if (t == 0) {
        float mx = simS[0]; int pred = 0;
        for (int k = 1; k < KSEL; ++k)
            if (simS[k] > mx) { mx = simS[k]; pred = k; }   // first-max tie break
        outMax[b] = mx;

        const int g = idx[b * KSEL + pred];
        gStar = g;

        const float* bx = boxes + (size_t)b * G * 15 + (size_t)g * 15;
        // anchor argmax on objectness (channel 4 is unchanged by cxcywh->xyxy)
        float bestObj = bx[4]; int aidx = 0;
        if (bx[9]  > bestObj) { bestObj = bx[9];  aidx = 1; }
        if (bx[14] > bestObj) {                   aidx = 2; }
        aStar = aidx;

        const float cx = bx[aidx * 5 + 0], cy = bx[aidx * 5 + 1];
        const float w  = bx[aidx * 5 + 2], h  = bx[aidx * 5 + 3];
        const float x1 = cx - w * 0.5f, y1 = cy - h * 0.5f;
        outBox[b * 5 + 0] = x1;
        outBox[b * 5 + 1] = y1;
        outBox[b * 5 + 2] = x1 + w;
        outBox[b * 5 + 3] = y1 + h;
        outBox[b * 5 + 4] = bx[aidx * 5 + 4];
    }
    __syncthreads();

    if (t < 32) {
        const float* mk = masks + ((size_t)b * G * 3 + (size_t)gStar * 3 + aStar) * 32;
        outMask[b * 32 + t] = mk[t];
    }
}

// ---------------------------------------------------------------------------
extern "C" void kernel_launch(void* const* d_in, const int* in_sizes, int n_in,
                              void* d_out, int out_size, void* d_ws, size_t ws_size,
                              hipStream_t stream)
{
    const float* boxes = (const float*)d_in[0];  // [256,169,3,5]
    const float* masks = (const float*)d_in[1];  // [256,169,3,32]
    const float* feat  = (const float*)d_in[2];  // [256,1024,13,13]
    const float* lang  = (const float*)d_in[3];  // [256,512]
    const float* W_vs  = (const float*)d_in[4];  // [1024,512]
    const float* b_vs  = (const float*)d_in[5];  // [512]
    const float* W_ts  = (const float*)d_in[6];  // [512,512]
    const float* b_ts  = (const float*)d_in[7];  // [512]
    // d_in[8] = select_num (== 32, compile-time constant here)

    const int bs = 256, D = 1024, H = 512;

    float* ws    = (float*)d_ws;
    float* y_new = ws;                            // 256*512
    float* vvec  = ws + bs * H;                   // 256*1024
    int*   idx   = (int*)(ws + bs * H + bs * D);  // 256*32 ints

    float* outBox  = (float*)d_out;               // 256*5
    float* outMask = outBox + bs * 5;             // 256*32
    float* outMax  = outMask + bs * 32;           // 256

    // 1) top-32 grids per batch
    topk_kernel<<<bs, 256, 0, stream>>>(boxes, idx);

    // 2) y_new = lang @ W_ts + b_ts        (M=256, N=512, K=512; B row-major)
    gemm_f32_wmma<0, 1><<<dim3(H / 16, bs / 16), 32, 0, stream>>>(
        lang, W_ts, y_new, b_ts, bs, H, H);

    // 3) v = y_new @ W_vs^T                (M=256, N=1024, K=512; B stored [N,K])
    gemm_f32_wmma<1, 0><<<dim3(D / 16, bs / 16), 32, 0, stream>>>(
        y_new, W_vs, vvec, nullptr, bs, D, H);

    // 4) fused sim + selection + outputs
    sim_select_kernel<<<bs, 256, 0, stream>>>(
        feat, vvec, y_new, b_vs, idx, boxes, masks, outBox, outMask, outMax);
}